// LSTM_29326036697973
// MI455X (gfx1250) — compile-verified
//
#include <hip/hip_runtime.h>
#include <math.h>

// ---------------------------------------------------------------------------
// 3-layer LSTM stack for MI455X (gfx1250), fp32 WMMA (V_WMMA_F32_16X16X4_F32).
// T=2048, I=64, H=256, H2=1024, 21 channels folded into batch for layer 1.
// ---------------------------------------------------------------------------

typedef __attribute__((ext_vector_type(2))) float v2f;
typedef __attribute__((ext_vector_type(8))) float v8f;

constexpr int TT  = 2048;   // time steps
constexpr int FI  = 64;     // layer-1 input feats
constexpr int NB1 = 21;     // layer-1 batch (channels)
constexpr int H1  = 256;
constexpr int G1  = 4 * H1;     // 1024
constexpr int H2  = 1024;
constexpr int G2  = 4 * H2;     // 4096
constexpr int KC1 = H1 + FI;    // 320: combined [h | x] K for layer-1 recurrence
constexpr int K2  = NB1 * H1;   // 5376: layer-2 input size
constexpr int L1NB = 8;         // blocks for layer-1 recurrence
constexpr int L1CELLS = NB1 * H1 / L1NB;  // 672 cells per block

__device__ __forceinline__ float sigf(float x) { return 1.0f / (1.0f + __expf(-x)); }

// ---------------------------------------------------------------------------
// CDNA5 async global->LDS copy (ASYNCcnt path).  Per-lane: copies 16 bytes
// from the lane's global address to the lane's LDS address.  Generic pointers
// to __shared__ objects carry the LDS byte address in their low 32 bits.
// ---------------------------------------------------------------------------
__device__ __forceinline__ void async_copy_b128(const float* gsrc, float* ldst) {
    unsigned l = (unsigned)(size_t)ldst;
    asm volatile("global_load_async_to_lds_b128 %0, %1, off"
                 :: "v"(l), "v"(gsrc) : "memory");
}
__device__ __forceinline__ void wait_async0() {
    asm volatile("s_wait_asynccnt 0x0" ::: "memory");
}

// ---------------------------------------------------------------------------
// Small helper kernels
// ---------------------------------------------------------------------------
__global__ void addv_k(const float* __restrict__ a, const float* __restrict__ b,
                       float* __restrict__ o, int n) {
    int i = blockIdx.x * 256 + threadIdx.x;
    if (i < n) o[i] = a[i] + b[i];
}

__global__ void zerov_k(float* __restrict__ p, int n) {
    int i = blockIdx.x * 256 + threadIdx.x;
    if (i < n) p[i] = 0.0f;
}

// out[c*R + r] = in[r*C + c]   (in: R x C row-major)
__global__ void transpose_k(const float* __restrict__ in, float* __restrict__ out,
                            int R, int C) {
    long i = (long)blockIdx.x * 256 + threadIdx.x;
    if (i < (long)R * C) {
        int r = (int)(i / C), c = (int)(i % C);
        out[(long)c * R + r] = in[i];
    }
}

// Pair-swizzle a weight matrix W[G1 rows][C cols] into fragment order:
// out[((kOff+k)>>1)*(2*G1) + 2*g + ((kOff+k)&1)] = W[g*C + k]
// so a WMMA B-fragment (values W[kb][g], W[kb+1][g]) is one aligned b64 load.
__global__ void swz_pairs_k(const float* __restrict__ W, float* __restrict__ out,
                            int C, int kOff) {
    long i = (long)blockIdx.x * 256 + threadIdx.x;
    if (i < (long)G1 * C) {
        int g = (int)(i / C), k = (int)(i % C);
        int kk = kOff + k;
        out[(size_t)(kk >> 1) * (2 * G1) + 2 * g + (kk & 1)] = W[i];
    }
}

// x [T][64][21]  ->  x1 [(t*21+b)][64]
__global__ void pack_x_k(const float* __restrict__ x, float* __restrict__ x1) {
    int i = blockIdx.x * 256 + threadIdx.x;
    if (i < TT * NB1 * FI) {
        int t = i / (NB1 * FI);
        int rem = i % (NB1 * FI);
        int b = rem >> 6, k = rem & 63;
        x1[i] = x[(t * FI + k) * NB1 + b];
    }
}

// ---------------------------------------------------------------------------
// Grid-wide barrier for persistent multi-block recurrence kernels.
// ---------------------------------------------------------------------------
__device__ __forceinline__ void grid_barrier(int* bar, int idx, int nb) {
    __syncthreads();
    if (threadIdx.x == 0) {
        __threadfence();
        atomicAdd(&bar[idx], 1);
        while (__hip_atomic_load(&bar[idx], __ATOMIC_ACQUIRE,
                                 __HIP_MEMORY_SCOPE_AGENT) < nb) {
            __builtin_amdgcn_s_sleep(1);
        }
        __threadfence();
    }
    __syncthreads();
}

// ---------------------------------------------------------------------------
// fp32 WMMA GEMM:  D[M][N] = A[M][K] * Bt[K][N] + bias[N]     (compile-time N,K)
// Block: 256 threads = 8 waves; block tile 64(M) x 128(N); 32x32 per wave
// (4 accumulators -> 4 WMMA per fragment-load group).  A (64x32) and B (32x128)
// chunks staged in LDS via async global->LDS b128 copies.
// Requires M%64==0, N%128==0, K%32==0.
// ---------------------------------------------------------------------------
template <int N, int K>
__global__ void __launch_bounds__(256)
wmma_gemm_k(const float* __restrict__ A, const float* __restrict__ Bt,
            const float* __restrict__ bias, float* __restrict__ D) {
    __shared__ float As[64 * 32];    // [m][k]
    __shared__ float Bs[32 * 128];   // [k][n]
    const int tid = threadIdx.x, lane = tid & 31, wave = tid >> 5;
    const int lrow = lane & 15, khalf = lane >> 4;
    const int mblk = blockIdx.x * 64, nblk = blockIdx.y * 128;
    const int mBase = (wave & 1) * 32;
    const int nBase = (wave >> 1) * 32;
    const int g0 = nblk + nBase + lrow;
    const int g1 = g0 + 16;

    v8f acc00, acc01, acc10, acc11;
    {
        float bv0 = bias[g0], bv1 = bias[g1];
#pragma unroll
        for (int r = 0; r < 8; ++r) {
            acc00[r] = bv0; acc01[r] = bv1;
            acc10[r] = bv0; acc11[r] = bv1;
        }
    }

    for (int kc = 0; kc < K; kc += 32) {
        __syncthreads();
        // stage A chunk: 64x32 = 512 float4, 2 per thread
#pragma unroll
        for (int j = 0; j < 2; ++j) {
            int i4 = tid + j * 256;
            int r = i4 >> 3, c = (i4 & 7) << 2;
            async_copy_b128(A + (size_t)(mblk + r) * K + kc + c, &As[r * 32 + c]);
        }
        // stage B chunk: 32x128 = 1024 float4, 4 per thread
#pragma unroll
        for (int j = 0; j < 4; ++j) {
            int i4 = tid + j * 256;
            int r = i4 >> 5, c = (i4 & 31) << 2;
            async_copy_b128(Bt + (size_t)(kc + r) * N + nblk + c, &Bs[r * 128 + c]);
        }
        wait_async0();
        __syncthreads();

#pragma unroll
        for (int k0 = 0; k0 < 32; k0 += 4) {
            int kb = k0 + (khalf << 1);
            v2f a0, a1, b0, b1;
            const float* ar0 = &As[(mBase + lrow) * 32];
            const float* ar1 = &As[(mBase + 16 + lrow) * 32];
            a0.x = ar0[kb];              a0.y = ar0[kb + 1];
            a1.x = ar1[kb];              a1.y = ar1[kb + 1];
            b0.x = Bs[kb * 128 + nBase + lrow];
            b0.y = Bs[(kb + 1) * 128 + nBase + lrow];
            b1.x = Bs[kb * 128 + nBase + 16 + lrow];
            b1.y = Bs[(kb + 1) * 128 + nBase + 16 + lrow];
            acc00 = __builtin_amdgcn_wmma_f32_16x16x4_f32(false, a0, false, b0, (short)0, acc00, false, false);
            acc01 = __builtin_amdgcn_wmma_f32_16x16x4_f32(false, a0, false, b1, (short)0, acc01, false, false);
            acc10 = __builtin_amdgcn_wmma_f32_16x16x4_f32(false, a1, false, b0, (short)0, acc10, false, false);
            acc11 = __builtin_amdgcn_wmma_f32_16x16x4_f32(false, a1, false, b1, (short)0, acc11, false, false);
        }
    }

#pragma unroll
    for (int r = 0; r < 8; ++r) {
        int m0 = mblk + mBase + r + (khalf << 3);
        D[(size_t)m0 * N + g0] = acc00[r];
        D[(size_t)m0 * N + g1] = acc01[r];
        D[(size_t)(m0 + 16) * N + g0] = acc10[r];
        D[(size_t)(m0 + 16) * N + g1] = acc11[r];
    }
}

// ---------------------------------------------------------------------------
// Layer-1 recurrence: persistent, 8 blocks x 256 threads, grid barrier / phase.
// Per step: Z[32 x 1024] = [h_prev | x_t](32 x 320) * W1c(320 x 1024) + b1.
// Each block owns a 128-column N slice (2 M-tiles x 8 N-tiles over 8 waves).
// W1sw is pair-swizzled so each B fragment is a single b64 load.
// zbuf has 32 rows so tile stores need no row predication (rows 21..31 dead).
// ---------------------------------------------------------------------------
__global__ void __launch_bounds__(256)
lstm1_rec_k(const float* __restrict__ x1, const float* __restrict__ W1sw,
            const float* __restrict__ b1, float* __restrict__ h1cat,
            float* __restrict__ zbuf,     // [32][1024]
            float* h_cur,                 // [21*256]
            float* c_cur,                 // [21*256]
            int* bar) {
    __shared__ float A_s[32 * KC1];   // rows 21..31 stay zero
    const int tid = threadIdx.x, lane = tid & 31, wave = tid >> 5;
    const int lrow = lane & 15, khalf = lane >> 4;
    const int nblk = blockIdx.x * 128;

    for (int i = tid; i < 32 * KC1; i += 256) A_s[i] = 0.0f;

    for (int t = 0; t < TT; ++t) {
        __syncthreads();
        // stage h_prev (21x256 = 1344 float4) into A_s rows 0..20, async b128
        for (int i4 = tid; i4 < NB1 * H1 / 4; i4 += 256) {
            int r = i4 >> 6, c = (i4 & 63) << 2;
            async_copy_b128(h_cur + r * H1 + c, &A_s[r * KC1 + c]);
        }
        // stage x_t (21x64 = 336 float4) into A_s cols 256..319
        for (int i4 = tid; i4 < NB1 * FI / 4; i4 += 256) {
            int r = i4 >> 4, c = (i4 & 15) << 2;
            async_copy_b128(x1 + (size_t)(t * NB1 + r) * FI + c,
                            &A_s[r * KC1 + H1 + c]);
        }
        wait_async0();
        __syncthreads();

        // 16 tiles (2 M x 8 N) over 8 waves -> 2 tiles/wave
#pragma unroll
        for (int rep = 0; rep < 2; ++rep) {
            int tt = wave + rep * 8;            // 0..15
            int m0 = (tt >> 3) << 4;            // 0 or 16
            int g  = nblk + ((tt & 7) << 4) + lrow;
            v8f acc;
            {
                float bv = b1[g];               // same bias for every row
#pragma unroll
                for (int r = 0; r < 8; ++r) acc[r] = bv;
            }
            const float* arow = &A_s[(m0 + lrow) * KC1];
            const float* wrow = &W1sw[2 * g];
#pragma unroll 4
            for (int k0 = 0; k0 < KC1; k0 += 4) {
                int kb = k0 + (khalf << 1);
                v2f a, b;
                a.x = arow[kb];
                a.y = arow[kb + 1];
                b = *(const v2f*)&wrow[(size_t)(kb >> 1) * (2 * G1)];
                acc = __builtin_amdgcn_wmma_f32_16x16x4_f32(
                    false, a, false, b, (short)0, acc, false, false);
            }
#pragma unroll
            for (int r = 0; r < 8; ++r) {
                int row = m0 + r + (khalf << 3);   // 0..31, rows >=21 are dead
                zbuf[row * G1 + g] = acc[r];
            }
        }
        grid_barrier(bar, 2 * t, L1NB);

        // gates: 672 cells per block
        for (int i = blockIdx.x * L1CELLS + tid;
             i < blockIdx.x * L1CELLS + L1CELLS; i += 256) {
            int bch = i >> 8, j = i & 255;
            const float* zr = &zbuf[bch * G1];
            float c = sigf(zr[H1 + j]) * c_cur[i] + sigf(zr[j]) * tanhf(zr[2 * H1 + j]);
            float h = sigf(zr[3 * H1 + j]) * tanhf(c);
            c_cur[i] = c;
            h_cur[i] = h;
            h1cat[(size_t)t * K2 + i] = h;   // channel-major concat
        }
        grid_barrier(bar, 2 * t + 1, L1NB);
    }
}

// ---------------------------------------------------------------------------
// Batch=1 LSTM recurrence (layers 2 & 3): persistent, NBLK blocks x 256.
// Phase A: thread g computes z[g] = xw[t][g] + h . WhhT[:,g]  (coalesced
//          WhhT streaming spread over NBLK WGPs, h staged in LDS).
// Phase B: G/4/NBLK cells per block update c/h.   Requires G == NBLK*256.
// ---------------------------------------------------------------------------
template <int HIN, int G, int NBLK>
__global__ void __launch_bounds__(256)
lstm_gemv_rec_k(const float* __restrict__ xw, const float* __restrict__ WhhT,
                float* __restrict__ h_all, float* h_cur, float* c_cur,
                float* zbuf, int* bar) {
    __shared__ float hsh[HIN];
    const int tid = threadIdx.x;
    const int g = blockIdx.x * 256 + tid;
    constexpr int Hc = G >> 2;
    constexpr int cpb = Hc / NBLK;

    for (int t = 0; t < TT; ++t) {
        for (int i = tid; i < HIN; i += 256) hsh[i] = h_cur[i];
        __syncthreads();

        float s = xw[(size_t)t * G + g];
        const float* wp = WhhT + g;
#pragma unroll 8
        for (int j = 0; j < HIN; ++j) s += hsh[j] * wp[(size_t)j * G];
        zbuf[g] = s;
        grid_barrier(bar, 2 * t, NBLK);

        if (tid < cpb) {
            int j = blockIdx.x * cpb + tid;
            float c = sigf(zbuf[Hc + j]) * c_cur[j] +
                      sigf(zbuf[j]) * tanhf(zbuf[2 * Hc + j]);
            float h = sigf(zbuf[3 * Hc + j]) * tanhf(c);
            c_cur[j] = c;
            h_cur[j] = h;
            h_all[(size_t)t * Hc + j] = h;
        }
        grid_barrier(bar, 2 * t + 1, NBLK);
        __syncthreads();
    }
}

// ---------------------------------------------------------------------------
// Final FC head: out[t][o] = bfc[o] + h3[t] . Wfc[o]   (21 outputs per t)
// ---------------------------------------------------------------------------
__global__ void __launch_bounds__(256)
fc_k(const float* __restrict__ h3, const float* __restrict__ Wfc,
     const float* __restrict__ bfc, float* __restrict__ out) {
    __shared__ float hb[H1];
    const int t = blockIdx.x, tid = threadIdx.x;
    hb[tid] = h3[(size_t)t * H1 + tid];
    __syncthreads();
    if (tid < 21) {
        float s = bfc[tid];
#pragma unroll 8
        for (int j = 0; j < H1; ++j) s += hb[j] * Wfc[tid * H1 + j];
        out[t * 21 + tid] = s;
    }
}

// ---------------------------------------------------------------------------
// Host-side launch
// ---------------------------------------------------------------------------
static inline int cdiv(long a, int b) { return (int)((a + b - 1) / b); }

extern "C" void kernel_launch(void* const* d_in, const int* in_sizes, int n_in,
                              void* d_out, int out_size, void* d_ws, size_t ws_size,
                              hipStream_t stream) {
    const float* x    = (const float*)d_in[0];
    const float* Wih1 = (const float*)d_in[1];
    const float* Whh1 = (const float*)d_in[2];
    const float* bih1 = (const float*)d_in[3];
    const float* bhh1 = (const float*)d_in[4];
    const float* Wih2 = (const float*)d_in[5];
    const float* Whh2 = (const float*)d_in[6];
    const float* bih2 = (const float*)d_in[7];
    const float* bhh2 = (const float*)d_in[8];
    const float* Wih3 = (const float*)d_in[9];
    const float* Whh3 = (const float*)d_in[10];
    const float* bih3 = (const float*)d_in[11];
    const float* bhh3 = (const float*)d_in[12];
    const float* Wfc  = (const float*)d_in[13];
    const float* bfc  = (const float*)d_in[14];
    float* out = (float*)d_out;

    // ---- workspace layout (floats) ----
    float* ws = (float*)d_ws;
    size_t o = 0;
    float* x1    = ws + o; o += (size_t)TT * NB1 * FI;   // packed x
    float* W1sw  = ws + o; o += (size_t)KC1 * G1;        // pair-swizzled [Whh1T;Wih1T]
    float* Wih2T = ws + o; o += (size_t)K2 * G2;
    float* Whh2T = ws + o; o += (size_t)H2 * G2;
    float* Wih3T = ws + o; o += (size_t)H2 * G1;
    float* Whh3T = ws + o; o += (size_t)H1 * G1;
    float* b1    = ws + o; o += G1;
    float* b2    = ws + o; o += G2;
    float* b3    = ws + o; o += G1;
    float* h1cat = ws + o; o += (size_t)TT * K2;
    float* xw2   = ws + o; o += (size_t)TT * G2;
    float* h2    = ws + o; o += (size_t)TT * H2;
    float* xw3   = ws + o; o += (size_t)TT * G1;
    float* h3    = ws + o; o += (size_t)TT * H1;
    float* state = ws + o;   // everything below is zero-initialized each launch
    float* z1   = state;                 // 32*1024 (rows 21..31 dead)
    float* hc1  = z1 + 32 * G1;          // 21*256
    float* c1   = hc1 + NB1 * H1;        // 21*256
    float* hc2  = c1 + NB1 * H1;         // 1024
    float* c2   = hc2 + H2;              // 1024
    float* z2   = c2 + H2;               // 4096
    float* hc3  = z2 + G2;               // 256
    float* c3   = hc3 + H1;              // 256
    float* z3   = c3 + H1;               // 1024
    int nstate = 32 * G1 + 2 * NB1 * H1 + 2 * H2 + G2 + 2 * H1 + G1;
    int* bar = (int*)(z3 + G1);
    int nbar = 3 * 2 * TT;               // three barrier arrays
    (void)ws_size; (void)in_sizes; (void)n_in; (void)out_size;

    // ---- 1. fold biases ----
    addv_k<<<cdiv(G1, 256), 256, 0, stream>>>(bih1, bhh1, b1, G1);
    addv_k<<<cdiv(G2, 256), 256, 0, stream>>>(bih2, bhh2, b2, G2);
    addv_k<<<cdiv(G1, 256), 256, 0, stream>>>(bih3, bhh3, b3, G1);

    // ---- 2. weight layout transforms ----
    swz_pairs_k<<<cdiv((long)G1 * H1, 256), 256, 0, stream>>>(Whh1, W1sw, H1, 0);
    swz_pairs_k<<<cdiv((long)G1 * FI, 256), 256, 0, stream>>>(Wih1, W1sw, FI, H1);
    transpose_k<<<cdiv((long)G2 * K2, 256), 256, 0, stream>>>(Wih2, Wih2T, G2, K2);
    transpose_k<<<cdiv((long)G2 * H2, 256), 256, 0, stream>>>(Whh2, Whh2T, G2, H2);
    transpose_k<<<cdiv((long)G1 * H2, 256), 256, 0, stream>>>(Wih3, Wih3T, G1, H2);
    transpose_k<<<cdiv((long)G1 * H1, 256), 256, 0, stream>>>(Whh3, Whh3T, G1, H1);

    // ---- 3. pack x to [T*21][64] ----
    pack_x_k<<<cdiv((long)TT * NB1 * FI, 256), 256, 0, stream>>>(x, x1);

    // ---- 4. zero recurrent state + grid barriers ----
    zerov_k<<<cdiv(nstate + nbar, 256), 256, 0, stream>>>(state, nstate + nbar);

    // ---- 5. layer-1 recurrence (WMMA, 8 blocks + grid barrier) ----
    lstm1_rec_k<<<L1NB, 256, 0, stream>>>(x1, W1sw, b1, h1cat, z1, hc1, c1, bar);

    // ---- 6. xw2 = h1cat[2048 x 5376] * Wih2T + b2  (90 GFLOP WMMA GEMM) ----
    wmma_gemm_k<G2, K2><<<dim3(TT / 64, G2 / 128), 256, 0, stream>>>(
        h1cat, Wih2T, b2, xw2);

    // ---- 7. layer-2 recurrence (16 blocks, grid barrier per phase) ----
    lstm_gemv_rec_k<H2, G2, G2 / 256><<<G2 / 256, 256, 0, stream>>>(
        xw2, Whh2T, h2, hc2, c2, z2, bar + 2 * TT);

    // ---- 8. xw3 = h2[2048 x 1024] * Wih3T + b3 ----
    wmma_gemm_k<G1, H2><<<dim3(TT / 64, G1 / 128), 256, 0, stream>>>(
        h2, Wih3T, b3, xw3);

    // ---- 9. layer-3 recurrence (4 blocks) ----
    lstm_gemv_rec_k<H1, G1, G1 / 256><<<G1 / 256, 256, 0, stream>>>(
        xw3, Whh3T, h3, hc3, c3, z3, bar + 4 * TT);

    // ---- 10. FC head ----
    fc_k<<<TT, 256, 0, stream>>>(h3, Wfc, bfc, out);
}